// NFNDoubleQuantizer_split_70360154243710
// MI455X (gfx1250) — compile-verified
//
#include <hip/hip_runtime.h>
#include <stdint.h>

// NF4 double-quantizer roundtrip for MI455X (gfx1250).
// Memory-bound: 64MB in + 64MB out => ~5.5us floor at 23.3 TB/s.
// Row (32KB) staged once in LDS via CDNA5 async global->LDS engine (ASYNCcnt),
// consumed twice (block stats pass + quantize/dequant pass).
// Codebook snap uses the table's odd symmetry: 7 compares on |xn| + sign restore.

#define ROWLEN   8192
#define NB       128       // 64-elem blocks per row
#define BS       64
#define NF_EPS   1e-8f
#define THREADS  256
#define NWAVES   (THREADS / 32)
#define BPW      (NB / NWAVES)   // 16 blocks per wave

__device__ __forceinline__ unsigned lds_off_u32(const void* p) {
    // Generic pointer to LDS: low 32 bits are the wave-relative LDS byte offset.
    return (unsigned)(uintptr_t)p;
}

__device__ __forceinline__ void async_copy_b128_to_lds(void* lds_dst, const void* gsrc) {
    unsigned loff = lds_off_u32(lds_dst);
    asm volatile("global_load_async_to_lds_b128 %0, %1, off"
                 :: "v"(loff), "v"(gsrc)
                 : "memory");
}

__device__ __forceinline__ void wait_asynccnt0() {
    asm volatile("s_wait_asynccnt 0" ::: "memory");
}

__global__ __launch_bounds__(THREADS)
void nf4_roundtrip_kernel(const float* __restrict__ x, float* __restrict__ out) {
    __shared__ float ldsData[ROWLEN];   // 32 KB row staging
    __shared__ float ldsMin[NB];
    __shared__ float ldsScale[NB];
    __shared__ float ldsS[2];           // row s_min, s_max

    const int row  = blockIdx.x;
    const int tid  = threadIdx.x;
    const int wave = tid >> 5;
    const int lane = tid & 31;

    // ---- Phase A: async DMA row -> LDS (2048 x b128, 8 per thread) ----
    const float* rowPtr = x + (size_t)row * ROWLEN;
    #pragma unroll
    for (int i = 0; i < (ROWLEN / 4) / THREADS; ++i) {
        int chunk = tid + i * THREADS;                 // float4 index
        async_copy_b128_to_lds(&ldsData[chunk * 4], rowPtr + chunk * 4);
    }
    wait_asynccnt0();
    __syncthreads();

    // ---- Positive half of NF4 codebook (odd-symmetric), rounded through fp16 ----
    const float tpraw[8] = {
        0.042096f, 0.127342f, 0.215949f, 0.310905f,
        0.416823f, 0.542209f, 0.707573f, 1.000000f };
    float tp[8];
    #pragma unroll
    for (int i = 0; i < 8; ++i) tp[i] = (float)(_Float16)tpraw[i];  // const-folded fp16 grid

    // ---- Phase B: per-block min/max (wave handles 16 blocks, b64 conflict-free) ----
    #pragma unroll
    for (int k = 0; k < BPW; ++k) {
        int b = wave * BPW + k;
        const float2 p = *reinterpret_cast<const float2*>(&ldsData[b * BS + 2 * lane]);
        float mn = fminf(p.x, p.y);
        float mx = fmaxf(p.x, p.y);
        #pragma unroll
        for (int off = 16; off >= 1; off >>= 1) {
            mn = fminf(mn, __shfl_xor(mn, off, 32));
            mx = fmaxf(mx, __shfl_xor(mx, off, 32));
        }
        if (lane == 0) {
            ldsMin[b]   = mn;
            ldsScale[b] = mx - mn;
        }
    }
    __syncthreads();

    // ---- Phase C: row-level min/max of the 128 scales (wave 0) ----
    if (tid < 32) {
        float s0 = ldsScale[tid];
        float s1 = ldsScale[tid + 32];
        float s2 = ldsScale[tid + 64];
        float s3 = ldsScale[tid + 96];
        float mn = fminf(fminf(s0, s1), fminf(s2, s3));
        float mx = fmaxf(fmaxf(s0, s1), fmaxf(s2, s3));
        #pragma unroll
        for (int off = 16; off >= 1; off >>= 1) {
            mn = fminf(mn, __shfl_xor(mn, off, 32));
            mx = fmaxf(mx, __shfl_xor(mx, off, 32));
        }
        if (tid == 0) { ldsS[0] = mn; ldsS[1] = mx; }
    }
    __syncthreads();

    const float smin = ldsS[0];
    const float sden = ldsS[1] - smin;           // s_max - s_min

    // ---- Phase D/E: quantize -> dequantize, store b64 ----
    const size_t outBase = (size_t)row * ROWLEN;
    #pragma unroll
    for (int k = 0; k < BPW; ++k) {
        int   b    = wave * BPW + k;
        float bmin = ldsMin[b];
        float s    = ldsScale[b];
        // double-quantized scale (jnp.round == rint, RTE)
        float sq   = rintf((s - smin) / (sden + NF_EPS) * 255.0f);
        float srec = smin + sq * (1.0f / 255.0f) * sden;
        float den  = s + NF_EPS;

        const int idx = b * BS + 2 * lane;
        const float2 p = *reinterpret_cast<const float2*>(&ldsData[idx]);

        float2 o;
        #pragma unroll
        for (int e = 0; e < 2; ++e) {
            float xv = (e == 0) ? p.x : p.y;
            // exact reference op order: 2*(x - min), / (s + eps), - 1
            float xn = (2.0f * (xv - bmin)) / den - 1.0f;
            float ax = fabsf(xn);   // folds into |src| modifier of v_cmp

            // nearest positive codebook level; strict '>' on midpoints
            float w = tp[0];
            #pragma unroll
            for (int i = 0; i < 7; ++i) {
                float mid = 0.5f * (tp[i] + tp[i + 1]);
                w = (ax > mid) ? tp[i + 1] : w;
            }
            // restore sign; xn==0 -> -tp[0] == t[7], matching argmin first-index
            w = (xn > 0.0f) ? w : -w;

            float r = (w + 1.0f) * 0.5f * srec + bmin;
            if (e == 0) o.x = r; else o.y = r;
        }
        *reinterpret_cast<float2*>(&out[outBase + idx]) = o;   // global_store_b64
    }
}

extern "C" void kernel_launch(void* const* d_in, const int* in_sizes, int n_in,
                              void* d_out, int out_size, void* d_ws, size_t ws_size,
                              hipStream_t stream) {
    (void)in_sizes; (void)n_in; (void)d_ws; (void)ws_size;
    const float* x = (const float*)d_in[0];
    float* out = (float*)d_out;
    int rows = out_size / ROWLEN;   // 2048 for the reference shape
    nf4_roundtrip_kernel<<<rows, THREADS, 0, stream>>>(x, out);
}